// AveragedGATConv_21612275434308
// MI455X (gfx1250) — compile-verified
//
#include <hip/hip_runtime.h>
#include <math.h>

typedef __attribute__((ext_vector_type(2))) float v2f;
typedef __attribute__((ext_vector_type(8))) float v8f;

#define IN_FEATS  128
#define OUT_FEATS 32
#define NUM_HEADS 8
#define HD        256   // NUM_HEADS*OUT_FEATS
#define NEG_SLOPE 0.2f

__device__ __forceinline__ unsigned enc_f32(float f) {
    unsigned u = __float_as_uint(f);
    return (u & 0x80000000u) ? ~u : (u | 0x80000000u);
}
__device__ __forceinline__ float dec_f32(unsigned e) {
    return (e & 0x80000000u) ? __uint_as_float(e & 0x7FFFFFFFu)
                             : __uint_as_float(~e);
}

// ---------------------------------------------------------------------------
// K1: ft = feat @ W  via V_WMMA_F32_16X16X4_F32 (fp32 matrix pipe, full prec).
// One wave -> 16x64 output strip (4 C fragments), K-loop in steps of 4.
// ---------------------------------------------------------------------------
__global__ __launch_bounds__(128) void gat_gemm_wmma(
    const float* __restrict__ feat, const float* __restrict__ W,
    float* __restrict__ ft, int n_nodes)
{
    int wave   = (blockIdx.x * blockDim.x + threadIdx.x) >> 5;
    int lane   = threadIdx.x & 31;
    int tile_m = wave >> 2;          // 16-row tile index
    int strip  = wave & 3;           // 64-col strip index (4 strips cover 256)
    int m0     = tile_m * 16;
    if (m0 >= n_nodes) return;       // uniform per wave

    int nlo  = lane & 15;            // N (or M) index within 16
    int half = lane >> 4;            // selects K pair {0,1} vs {2,3}
    int arow = m0 + nlo; if (arow >= n_nodes) arow = n_nodes - 1;  // clamp (M%16==0 normally)

    const int col0 = strip * 64 + nlo;

    v8f c0 = {}, c1 = {}, c2 = {}, c3 = {};

    #pragma unroll 4
    for (int k0 = 0; k0 < IN_FEATS; k0 += 4) {
        // A fragment: 16x4, 2 consecutive K values per lane
        v2f a = *(const v2f*)(feat + (size_t)arow * IN_FEATS + k0 + half * 2);
        // B fragments: 4x16 each, row-major W; lanes stripe N, half selects K pair
        const float* wp0 = W + (size_t)(k0 + half * 2) * HD;
        const float* wp1 = wp0 + HD;
        v2f b0 = { wp0[col0],      wp1[col0]      };
        v2f b1 = { wp0[col0 + 16], wp1[col0 + 16] };
        v2f b2 = { wp0[col0 + 32], wp1[col0 + 32] };
        v2f b3 = { wp0[col0 + 48], wp1[col0 + 48] };
        c0 = __builtin_amdgcn_wmma_f32_16x16x4_f32(false, a, false, b0, (short)0, c0, false, false);
        c1 = __builtin_amdgcn_wmma_f32_16x16x4_f32(false, a, false, b1, (short)0, c1, false, false);
        c2 = __builtin_amdgcn_wmma_f32_16x16x4_f32(false, a, false, b2, (short)0, c2, false, false);
        c3 = __builtin_amdgcn_wmma_f32_16x16x4_f32(false, a, false, b3, (short)0, c3, false, false);
    }

    // C/D layout: VGPR j -> M = j (lanes 0-15) / j+8 (lanes 16-31), N = lane&15
    #pragma unroll
    for (int j = 0; j < 8; ++j) {
        int r = m0 + half * 8 + j;
        if (r < n_nodes) {
            float* o = ft + (size_t)r * HD + strip * 64 + nlo;
            o[0]  = c0[j];
            o[16] = c1[j];
            o[32] = c2[j];
            o[48] = c3[j];
        }
    }
}

// ---------------------------------------------------------------------------
// K2: per (node, head): el/er dot products; init emax(-inf enc), esum(0);
//     init output h with mean-over-heads of bias.
// ---------------------------------------------------------------------------
__global__ __launch_bounds__(256) void gat_node_prep(
    const float* __restrict__ ft, const float* __restrict__ attn_l,
    const float* __restrict__ attn_r, const float* __restrict__ bias,
    float* __restrict__ el, float* __restrict__ er,
    unsigned* __restrict__ emax_enc, float* __restrict__ esum,
    float* __restrict__ hout, int n_nodes)
{
    int i = blockIdx.x * blockDim.x + threadIdx.x;
    if (i >= n_nodes * NUM_HEADS) return;
    int n = i >> 3, h = i & 7;

    const float* f  = ft + (size_t)n * HD + h * OUT_FEATS;
    const float* al = attn_l + h * OUT_FEATS;
    const float* ar = attn_r + h * OUT_FEATS;
    float sl = 0.f, sr = 0.f;
    #pragma unroll
    for (int d = 0; d < OUT_FEATS; ++d) { float v = f[d]; sl += v * al[d]; sr += v * ar[d]; }
    el[i] = sl;
    er[i] = sr;
    esum[i] = 0.f;
    emax_enc[i] = enc_f32(-INFINITY);

    // this (n,h) thread initializes h[n, h*4 .. h*4+3] with bias head-mean
    #pragma unroll
    for (int j = 0; j < 4; ++j) {
        int d = h * 4 + j;
        float acc = 0.f;
        #pragma unroll
        for (int hh = 0; hh < NUM_HEADS; ++hh) acc += bias[hh * OUT_FEATS + d];
        hout[(size_t)n * OUT_FEATS + d] = acc * 0.125f;
    }
}

// ---------------------------------------------------------------------------
// K3: segment-max of leaky_relu(el[src]+er[dst]) via order-preserving uint max
// ---------------------------------------------------------------------------
__global__ __launch_bounds__(256) void gat_edge_max(
    const int* __restrict__ src, const int* __restrict__ dst,
    const float* __restrict__ el, const float* __restrict__ er,
    unsigned* __restrict__ emax_enc, int n_edges)
{
    int i = blockIdx.x * blockDim.x + threadIdx.x;
    if (i >= n_edges * NUM_HEADS) return;
    int e = i >> 3, h = i & 7;
    int s = src[e], d = dst[e];
    float x = el[s * NUM_HEADS + h] + er[d * NUM_HEADS + h];
    x = x > 0.f ? x : NEG_SLOPE * x;
    atomicMax(&emax_enc[d * NUM_HEADS + h], enc_f32(x));
}

// ---------------------------------------------------------------------------
// K4: segment-sum of exp(logit - max); logit recomputed (L2-resident gathers)
// ---------------------------------------------------------------------------
__global__ __launch_bounds__(256) void gat_edge_sum(
    const int* __restrict__ src, const int* __restrict__ dst,
    const float* __restrict__ el, const float* __restrict__ er,
    const unsigned* __restrict__ emax_enc, float* __restrict__ esum, int n_edges)
{
    int i = blockIdx.x * blockDim.x + threadIdx.x;
    if (i >= n_edges * NUM_HEADS) return;
    int e = i >> 3, h = i & 7;
    int s = src[e], d = dst[e];
    float x = el[s * NUM_HEADS + h] + er[d * NUM_HEADS + h];
    x = x > 0.f ? x : NEG_SLOPE * x;
    float m = dec_f32(emax_enc[d * NUM_HEADS + h]);
    atomicAdd(&esum[d * NUM_HEADS + h], __expf(x - m));
}

// ---------------------------------------------------------------------------
// K5: one wave per edge: a = softmax coeff; write attn; scatter
//     h[dst,d] += ft[src,h,d] * a / 8  (head-mean folded into the atomics).
//     lane -> head = lane/4, 8 feats starting at lane*8 within the 256.
// ---------------------------------------------------------------------------
__global__ __launch_bounds__(256) void gat_edge_msg(
    const int* __restrict__ src, const int* __restrict__ dst,
    const float* __restrict__ el, const float* __restrict__ er,
    const unsigned* __restrict__ emax_enc, const float* __restrict__ esum,
    const float* __restrict__ ft, float* __restrict__ hout,
    float* __restrict__ attn_out, int n_edges)
{
    int wave = (blockIdx.x * blockDim.x + threadIdx.x) >> 5;
    int lane = threadIdx.x & 31;
    if (wave >= n_edges) return;
    int e = wave;
    int s = src[e], d = dst[e];
    int h = lane >> 2;

    float x = el[s * NUM_HEADS + h] + er[d * NUM_HEADS + h];
    x = x > 0.f ? x : NEG_SLOPE * x;
    float m   = dec_f32(emax_enc[d * NUM_HEADS + h]);
    float a   = __expf(x - m) / esum[d * NUM_HEADS + h];

    if ((lane & 3) == 0) attn_out[(size_t)e * NUM_HEADS + h] = a;

    int fbase = lane * 8;                     // 0..255, head = fbase/32
    const float4* fp = (const float4*)(ft + (size_t)s * HD + fbase);
    float4 v0 = fp[0], v1 = fp[1];
    float sc = a * 0.125f;                    // fold mean over heads

    float* hp = hout + (size_t)d * OUT_FEATS + (lane & 3) * 8;
    atomicAdd(&hp[0], v0.x * sc);
    atomicAdd(&hp[1], v0.y * sc);
    atomicAdd(&hp[2], v0.z * sc);
    atomicAdd(&hp[3], v0.w * sc);
    atomicAdd(&hp[4], v1.x * sc);
    atomicAdd(&hp[5], v1.y * sc);
    atomicAdd(&hp[6], v1.z * sc);
    atomicAdd(&hp[7], v1.w * sc);
}

// ---------------------------------------------------------------------------
extern "C" void kernel_launch(void* const* d_in, const int* in_sizes, int n_in,
                              void* d_out, int out_size, void* d_ws, size_t ws_size,
                              hipStream_t stream)
{
    const float* feat   = (const float*)d_in[0];
    const int*   src    = (const int*)  d_in[1];
    const int*   dst    = (const int*)  d_in[2];
    const float* W      = (const float*)d_in[3];
    const float* attn_l = (const float*)d_in[4];
    const float* attn_r = (const float*)d_in[5];
    const float* bias   = (const float*)d_in[6];

    const int n_nodes = in_sizes[0] / IN_FEATS;
    const int n_edges = in_sizes[1];

    // workspace layout
    float*    ft   = (float*)d_ws;                               // n_nodes*HD
    float*    el   = ft + (size_t)n_nodes * HD;                  // n_nodes*8
    float*    er   = el + (size_t)n_nodes * NUM_HEADS;           // n_nodes*8
    unsigned* emax = (unsigned*)(er + (size_t)n_nodes * NUM_HEADS);
    float*    esum = (float*)(emax + (size_t)n_nodes * NUM_HEADS);

    // output layout: [h: n_nodes*32][attn: n_edges*8]
    float* hout     = (float*)d_out;
    float* attn_out = hout + (size_t)n_nodes * OUT_FEATS;

    // K1: WMMA projection GEMM
    {
        int mtiles = (n_nodes + 15) / 16;        // one block = 4 waves = 1 m-tile
        gat_gemm_wmma<<<mtiles, 128, 0, stream>>>(feat, W, ft, n_nodes);
    }
    // K2: el/er + inits
    {
        int total = n_nodes * NUM_HEADS;
        gat_node_prep<<<(total + 255) / 256, 256, 0, stream>>>(
            ft, attn_l, attn_r, bias, el, er, emax, esum, hout, n_nodes);
    }
    // K3: segment max
    {
        int total = n_edges * NUM_HEADS;
        gat_edge_max<<<(total + 255) / 256, 256, 0, stream>>>(src, dst, el, er, emax, n_edges);
    }
    // K4: segment sum of exp
    {
        int total = n_edges * NUM_HEADS;
        gat_edge_sum<<<(total + 255) / 256, 256, 0, stream>>>(src, dst, el, er, emax, esum, n_edges);
    }
    // K5: normalize + message scatter (one wave per edge)
    {
        int blocks = (n_edges + 7) / 8;          // 8 waves / 256-thread block
        gat_edge_msg<<<blocks, 256, 0, stream>>>(src, dst, el, er, emax, esum,
                                                 ft, hout, attn_out, n_edges);
    }
}